// Model_pro_52742198395334
// MI455X (gfx1250) — compile-verified
//
#include <hip/hip_runtime.h>
#include <hip/hip_bf16.h>

// CDNA5 / gfx1250: wave32, V_WMMA_F32_16X16X4_F32.
typedef __attribute__((ext_vector_type(2))) float v2f;
typedef __attribute__((ext_vector_type(8))) float v8f;

#define NATOM 2048
#define NLIST (NATOM + 4)      // + sentinel pad to a multiple of 4
#define NG    48
#define NG3   (NG * NG * NG)   // 110592
#define NC    16

// Per-atom derived data (identical math to the reference weight construction).
__device__ __forceinline__ void atom_data(const float* __restrict__ coords,
                                          const int* __restrict__ ch_idx,
                                          const int* __restrict__ fg,
                                          int i, float& X, float& Y, float& Z,
                                          float& ir, float& cut2, int& m) {
    X = coords[3 * i + 0];
    Y = coords[3 * i + 1];
    Z = coords[3 * i + 2];
    const int ch = ch_idx[i] & 3;
    // VDW radii: 1.52, 1.70, 1.55, 1.80
    const float r = (ch == 0) ? 1.52f : (ch == 1) ? 1.70f : (ch == 2) ? 1.55f : 1.80f;
    ir = 1.0f / r;
    const float rc = 1.5f * r;
    cut2 = rc * rc;
    const int f = fg[i];
    int c1 = (f == 14) ? 4 : ((f == 15) ? 6 : (f + 4));
    if (f == 12 || f == 13) c1 = -1;
    int c2 = (f == 14) ? 5 : ((f == 15) ? 9 : -1);
    m = (1 << ch);
    if (c1 >= 0) m |= (1 << c1);
    if (c2 >= 0) m |= (1 << c2);
}

// Block = one (x, y-run-of-8, z-tile) slab: 8 wave32 tiles of 16(channel)x16(z).
//   D[c][z] = sum_k w[k][c] * val[k][z]  via V_WMMA_F32_16X16X4_F32, K=4/step.
// Phase 1: deterministic (order-preserving, scan-based) compaction of the 2048
//          atoms against the block bbox + per-atom cutoff -> ~5% survive.
// Phase 2: WMMA loop over survivors with wave-uniform ballot skip per 4-atom
//          group (EXEC stays all-ones, as WMMA requires).
__global__ __launch_bounds__(256) void densgrid_wmma_kernel(
    const float* __restrict__ coords,   // [2048][3]
    const int*   __restrict__ ch_idx,   // [2048]
    const int*   __restrict__ fg,       // [2048]
    float*       __restrict__ out)      // [16][48][48][48]
{
    __shared__ float cx[NLIST];
    __shared__ float cy[NLIST];
    __shared__ float cz[NLIST];
    __shared__ float ccut[NLIST];   // (1.5*r)^2
    __shared__ float cir[NLIST];    // 1/r
    __shared__ int   cmask[NLIST];  // 16-bit channel one-hot-sum mask
    __shared__ int   scnt[256];     // scan buffer

    const int tid = threadIdx.x;

    // ---- Block geometry: 864 blocks = 3(zt) * 48(x) * 6(y-group) ----
    const int b   = blockIdx.x;
    const int ztb = b / 288;          // z tile: 0..2
    const int p   = b % 288;
    const int xb  = p / 6;            // 0..47
    const int yg  = p % 6;            // y = yg*8 + wave
    const float bx0 = 0.5f * (float)xb,            bx1 = bx0;
    const float by0 = 0.5f * (float)(yg * 8),      by1 = 0.5f * (float)(yg * 8 + 7);
    const float bz0 = 0.5f * (float)(ztb * 16),    bz1 = 0.5f * (float)(ztb * 16 + 15);

    // ---- Phase 1a: cull 8 consecutive atoms per thread against bbox+cutoff ----
    int keepbits = 0, cnt = 0;
    for (int u = 0; u < 8; ++u) {
        const int i = tid * 8 + u;
        float X, Y, Z, ir, cut2; int m;
        atom_data(coords, ch_idx, fg, i, X, Y, Z, ir, cut2, m);
        const float ddx = fmaxf(bx0 - X, fmaxf(X - bx1, 0.0f));
        const float ddy = fmaxf(by0 - Y, fmaxf(Y - by1, 0.0f));
        const float ddz = fmaxf(bz0 - Z, fmaxf(Z - bz1, 0.0f));
        const float d2b = ddx * ddx + ddy * ddy + ddz * ddz;
        if (d2b < cut2) { keepbits |= (1 << u); ++cnt; }
    }
    scnt[tid] = cnt;
    __syncthreads();
    // ---- Phase 1b: Hillis-Steele inclusive scan (order-preserving offsets) ----
    for (int off = 1; off < 256; off <<= 1) {
        const int v   = scnt[tid];
        const int add = (tid >= off) ? scnt[tid - off] : 0;
        __syncthreads();
        scnt[tid] = v + add;
        __syncthreads();
    }
    int o = scnt[tid] - cnt;          // exclusive offset; preserves atom order
    const int M = scnt[255];          // survivor count (uniform)
    // ---- Phase 1c: write compacted atom data (deterministic order) ----
    for (int u = 0; u < 8; ++u) {
        if (keepbits & (1 << u)) {
            const int i = tid * 8 + u;
            float X, Y, Z, ir, cut2; int m;
            atom_data(coords, ch_idx, fg, i, X, Y, Z, ir, cut2, m);
            cx[o] = X; cy[o] = Y; cz[o] = Z;
            cir[o] = ir; ccut[o] = cut2; cmask[o] = m;
            ++o;
        }
    }
    if (tid < 4) {                    // far sentinels: contribute exactly 0
        const int o2 = M + tid;
        cx[o2] = 1.0e6f; cy[o2] = 1.0e6f; cz[o2] = 1.0e6f;
        cir[o2] = 1.0f; ccut[o2] = -1.0f; cmask[o2] = 0;
    }
    __syncthreads();
    const int Mr = (M + 3) & ~3;      // padded, uniform loop bound

    // ---- Phase 2: tile / lane geometry ----
    const int lane = tid & 31;
    const int wave = tid >> 5;
    const int yw   = yg * 8 + wave;                // this wave's y
    const int n    = lane & 15;                    // N index: z offset / channel
    const int khalf = (lane >> 4) << 1;            // K base per lane half (0 or 2)
    const float gx = bx0;
    const float gy = 0.5f * (float)yw;
    const float gz = 0.5f * (float)(ztb * 16 + n);

    const float kNeg2Log2e = -2.0f * 1.44269504088896340736f;  // exp(t)=exp2(t*log2e)
    const float invE2      = 0.13533528323661270231f;          // 1/e^2

    v8f acc = {0.f, 0.f, 0.f, 0.f, 0.f, 0.f, 0.f, 0.f};

    for (int kb = 0; kb < Mr; kb += 4) {
        const int ka0 = kb + khalf;   // this lane's atoms (VGPR j = 0,1)
        const int ka1 = ka0 + 1;

        const float dx0 = cx[ka0] - gx, dy0 = cy[ka0] - gy, dz0 = cz[ka0] - gz;
        const float dx1 = cx[ka1] - gx, dy1 = cy[ka1] - gy, dz1 = cz[ka1] - gz;
        const float d20 = dx0 * dx0 + dy0 * dy0 + dz0 * dz0;
        const float d21 = dx1 * dx1 + dy1 * dy1 + dz1 * dz1;
        const bool near = (d20 < ccut[ka0]) | (d21 < ccut[ka1]);

        if (__builtin_amdgcn_ballot_w32(near)) {
            v2f a, bm;
            {   // j = 0
                const float q  = __builtin_amdgcn_sqrtf(d20) * cir[ka0];
                const float q2 = q * q;
                const float f1 = __builtin_amdgcn_exp2f(kNeg2Log2e * q2);
                const float f2 = (4.0f * q2 - 12.0f * q + 9.0f) * invE2;
                const float w1 = (q < 1.0f) ? 1.0f : 0.0f;
                const float w2 = (q < 1.5f) ? (1.0f - w1) : 0.0f;
                bm[0] = w1 * f1 + w2 * f2;
                a[0]  = (float)((cmask[ka0] >> n) & 1);
            }
            {   // j = 1
                const float q  = __builtin_amdgcn_sqrtf(d21) * cir[ka1];
                const float q2 = q * q;
                const float f1 = __builtin_amdgcn_exp2f(kNeg2Log2e * q2);
                const float f2 = (4.0f * q2 - 12.0f * q + 9.0f) * invE2;
                const float w1 = (q < 1.0f) ? 1.0f : 0.0f;
                const float w2 = (q < 1.5f) ? (1.0f - w1) : 0.0f;
                bm[1] = w1 * f1 + w2 * f2;
                a[1]  = (float)((cmask[ka1] >> n) & 1);
            }
            acc = __builtin_amdgcn_wmma_f32_16x16x4_f32(
                /*neg_a=*/false, a, /*neg_b=*/false, bm,
                /*c_mod=*/(short)0, acc, /*reuse_a=*/false, /*reuse_b=*/false);
        }
    }

    // ---- Store C/D tile: lane l, vgpr v -> channel c = v + 8*(l>>4), point n ----
    const size_t pbase = (size_t)xb * (NG * NG) + (size_t)yw * NG + (size_t)(ztb * 16 + n);
    const int    chalf = (lane >> 4) * 8;
#pragma unroll
    for (int v = 0; v < 8; ++v) {
        out[(size_t)(chalf + v) * NG3 + pbase] = acc[v];
    }
}

extern "C" void kernel_launch(void* const* d_in, const int* in_sizes, int n_in,
                              void* d_out, int out_size, void* d_ws, size_t ws_size,
                              hipStream_t stream) {
    (void)in_sizes; (void)n_in; (void)d_ws; (void)ws_size; (void)out_size;
    const float* coords = (const float*)d_in[0];
    const int*   ch     = (const int*)d_in[1];
    const int*   fgrp   = (const int*)d_in[2];
    float*       out    = (float*)d_out;

    // 864 blocks = 3 z-tiles * 48 x * 6 y-groups; 8 wave32 tiles per block.
    densgrid_wmma_kernel<<<864, 256, 0, stream>>>(coords, ch, fgrp, out);
}